// ModelNew_14448269984255
// MI455X (gfx1250) — compile-verified
//
#include <hip/hip_runtime.h>
#include <stdint.h>

// take_along_axis gather:
//   out[b, r, c] = x[b, r, wrap(index[b, r, c])]
//   x:     (8, 4096, 4096) f32   (only rows 0..2047 of axis 1 are used)
//   index: (8, 2048, 4096) i32   values in [-4096, 4096)
//   out:   (8, 2048, 4096) f32
//
// Strategy (MI455X, gfx1250):
//   - 1 workgroup per output row (16384 blocks x 256 threads, wave32).
//   - Stage the 16 KB x-row into LDS with the CDNA5 async copy engine
//     (global_load_async_to_lds_b128, ASYNCcnt), then gather from LDS
//     (ds_load_b32) so random 4B reads never hit HBM / cachelines.
//   - Index stream and output are fully coalesced b128 accesses.
//   - HBM-bound: ~768 MB total traffic -> ~33 us at 23.3 TB/s.

namespace {
constexpr int NB     = 8;      // batches
constexpr int XROWS  = 4096;   // x rows per batch
constexpr int IROWS  = 2048;   // index/out rows per batch
constexpr int N      = 4096;   // gather-dim extent
constexpr int BLOCK  = 256;    // threads per block (8 wave32)
constexpr int CHUNKS = N / (BLOCK * 4);  // 4: int4/float4 chunks per thread
}  // namespace

__global__ __launch_bounds__(BLOCK)
void gather_rows_lds_async(const float* __restrict__ x,
                           const int*  __restrict__ index,
                           float* __restrict__ out)
{
    __shared__ float srow[N];   // 16 KB staging buffer for one x row

    const int row = blockIdx.x;            // 0 .. NB*IROWS-1
    const int b   = row >> 11;             // row / IROWS
    const int r   = row & (IROWS - 1);     // row % IROWS

    const float* __restrict__ xrow = x     + ((size_t)b * XROWS + (size_t)r) * N;
    const int*   __restrict__ irow = index + (size_t)row * N;
    float*       __restrict__ orow = out   + (size_t)row * N;

    const int tid = threadIdx.x;

    // LDS byte address of srow: flat pointers into the LDS aperture carry the
    // LDS byte offset in their low 32 bits (CDNA5 ISA 10.2 aperture mapping).
    const unsigned lds_base =
        (unsigned)(unsigned long long)(uintptr_t)(void*)&srow[0];

    // ---- Phase 1: async copy x row (16 KB) into LDS ------------------------
    // Each lane moves 16 B per issue; 256 threads x 4 iters covers 16 KB.
    // GVS addressing: mem = SADDR(64b base) + VADDR(32b byte offset).
#pragma unroll
    for (int it = 0; it < CHUNKS; ++it) {
        const unsigned byte_off = (unsigned)(tid * 16 + it * (BLOCK * 16));
        const unsigned lds_addr = lds_base + byte_off;
        asm volatile(
            "global_load_async_to_lds_b128 %0, %1, %2"
            :
            : "v"(lds_addr), "v"(byte_off), "s"(xrow)
            : "memory");
    }
    // Wait for this wave's async transfers, then make them visible to all
    // waves in the workgroup.
    asm volatile("s_wait_asynccnt 0x0" ::: "memory");
    __syncthreads();

    // ---- Phase 2: vectorized index stream -> LDS gather -> vector store ----
#pragma unroll
    for (int it = 0; it < CHUNKS; ++it) {
        const int vid = tid + it * BLOCK;            // int4 / float4 chunk id
        int4 iv = reinterpret_cast<const int4*>(irow)[vid];

        // Negative-index wraparound (reference: where(idx < 0, idx + N, idx)).
        if (iv.x < 0) iv.x += N;
        if (iv.y < 0) iv.y += N;
        if (iv.z < 0) iv.z += N;
        if (iv.w < 0) iv.w += N;

        float4 ov;
        ov.x = srow[iv.x];   // ds_load_b32 gathers within the 16 KB row
        ov.y = srow[iv.y];
        ov.z = srow[iv.z];
        ov.w = srow[iv.w];

        reinterpret_cast<float4*>(orow)[vid] = ov;   // global_store_b128
    }
}

extern "C" void kernel_launch(void* const* d_in, const int* in_sizes, int n_in,
                              void* d_out, int out_size, void* d_ws, size_t ws_size,
                              hipStream_t stream) {
    (void)in_sizes; (void)n_in; (void)out_size; (void)d_ws; (void)ws_size;

    const float* x     = (const float*)d_in[0];
    const int*   index = (const int*)d_in[1];
    // d_in[2] is the python scalar dim (== 2); semantics are baked in above.
    float* out = (float*)d_out;

    dim3 grid(NB * IROWS);   // 16384 blocks, one per output row
    dim3 block(BLOCK);
    gather_rows_lds_async<<<grid, block, 0, stream>>>(x, index, out);
}